// featureGCN_28089086116689
// MI455X (gfx1250) — compile-verified
//
#include <hip/hip_runtime.h>
#include <hip/hip_bf16.h>
#include <math.h>

typedef __attribute__((ext_vector_type(2))) float v2f;
typedef __attribute__((ext_vector_type(8))) float v8f;

#define D_IN  128
#define D_H1  64
#define D_H2  16

// ---------------------------------------------------------------------------
// Dual fp32 WMMA GEMM: P = A @ Wl, T = A @ Wr
//   A : [n_nodes][K]   row-major
//   Wl, Wr : [K][DOUT] row-major   (reference stores weights as (in, out))
// One wave computes one 16x16 output tile for BOTH products, sharing A loads.
// Uses V_WMMA_F32_16X16X4_F32 (CDNA5 fp32 tensor op) — exact fp32 math.
//
// VGPR layouts per CDNA5 ISA 7.12.2:
//   A 16x4 f32 : vgpr v, lane = hi*16 + m  ->  A[m][k0 + v + 2*hi]
//   B 4x16 f32 : vgpr v, lane = hi*16 + n  ->  B[k0 + v + 2*hi][n]
//   C/D 16x16  : vgpr r, lane = hi*16 + n  ->  D[r + 8*hi][n]
// ---------------------------------------------------------------------------
template <int K, int DOUT>
__global__ void gemm_dual_wmma(const float* __restrict__ A,
                               const float* __restrict__ Wl,
                               const float* __restrict__ Wr,
                               float* __restrict__ P,
                               float* __restrict__ T,
                               int n_nodes, int nRowTiles) {
    const int tilesCol = DOUT / 16;
    const int wid  = (int)((blockIdx.x * blockDim.x + threadIdx.x) >> 5);
    const int lane = (int)(threadIdx.x & 31);
    const int total = nRowTiles * tilesCol;
    if (wid >= total) return;                 // wave-uniform: EXEC stays all-ones

    const int rt = wid / tilesCol;
    const int ct = wid - rt * tilesCol;
    const int m  = lane & 15;
    const int hi = lane >> 4;

    // clamp A row for tail tiles (keeps EXEC full; stores handled below)
    int arowIdx = rt * 16 + m;
    if (arowIdx >= n_nodes) arowIdx = n_nodes - 1;

    const float* __restrict__ ap = A  + (size_t)arowIdx * K + 2 * hi;
    const float* __restrict__ bl = Wl + (size_t)(2 * hi) * DOUT + ct * 16 + m;
    const float* __restrict__ br = Wr + (size_t)(2 * hi) * DOUT + ct * 16 + m;

    v8f cl = {};
    v8f cr = {};

    #pragma unroll 4
    for (int k0 = 0; k0 < K; k0 += 4) {
        const float2 a2 = *(const float2*)ap;        // A[m][k0+2hi], A[m][k0+2hi+1]
        v2f av; av.x = a2.x; av.y = a2.y;

        v2f bv; bv.x = bl[0]; bv.y = bl[DOUT];       // Wl[k0+2hi][col], Wl[k0+2hi+1][col]
        cl = __builtin_amdgcn_wmma_f32_16x16x4_f32(
                 false, av, false, bv, (short)0, cl, false, false);

        v2f cv; cv.x = br[0]; cv.y = br[DOUT];
        cr = __builtin_amdgcn_wmma_f32_16x16x4_f32(
                 false, av, false, cv, (short)0, cr, false, false);

        ap += 4;
        bl += 4 * DOUT;
        br += 4 * DOUT;
    }

    const int col = ct * 16 + m;
    float* __restrict__ prow = P + (size_t)(rt * 16 + 8 * hi) * DOUT + col;
    float* __restrict__ trow = T + (size_t)(rt * 16 + 8 * hi) * DOUT + col;

    if (rt * 16 + 16 <= n_nodes) {
        // wave-uniform fast path: whole tile in bounds -> plain strided stores
        #pragma unroll
        for (int r = 0; r < 8; ++r) {
            prow[(size_t)r * DOUT] = cl[r];
            trow[(size_t)r * DOUT] = cr[r];
        }
    } else {
        // cold tail tile (unused at N=100000 but kept for generality)
        #pragma unroll
        for (int r = 0; r < 8; ++r) {
            const int row = rt * 16 + r + 8 * hi;
            if (row < n_nodes) {
                prow[(size_t)r * DOUT] = cl[r];
                trow[(size_t)r * DOUT] = cr[r];
            }
        }
    }
}

// ---------------------------------------------------------------------------
// Scatter-add of transformed messages along edges (L2-resident atomics),
// fused degree accumulation for the 64-feature layer.
// ---------------------------------------------------------------------------
__global__ void scatter_add64(const float* __restrict__ p,
                              const long long* __restrict__ src,
                              const long long* __restrict__ dst,
                              float* __restrict__ agg,
                              float* __restrict__ deg,
                              int n_edges) {
    const int i = (int)(blockIdx.x * blockDim.x + threadIdx.x);
    if (i >= n_edges * 64) return;
    const int e = i >> 6;
    const int f = i & 63;
    const int s = (int)src[e];
    const int d = (int)dst[e];
    atomicAdd(&agg[(size_t)d * 64 + f], p[(size_t)s * 64 + f]);
    if (f == 0) atomicAdd(&deg[d], 1.0f);
}

__global__ void scatter_add16(const float* __restrict__ p,
                              const long long* __restrict__ src,
                              const long long* __restrict__ dst,
                              float* __restrict__ agg,
                              int n_edges) {
    const int i = (int)(blockIdx.x * blockDim.x + threadIdx.x);
    if (i >= n_edges * 16) return;
    const int e = i >> 4;
    const int f = i & 15;
    const int s = (int)src[e];
    const int d = (int)dst[e];
    atomicAdd(&agg[(size_t)d * 16 + f], p[(size_t)s * 16 + f]);
}

// h = agg/max(deg,1) + t + b   (written in place over agg)
template <int D>
__global__ void combine(float* __restrict__ aggh,
                        const float* __restrict__ t,
                        const float* __restrict__ deg,
                        const float* __restrict__ b,
                        int n_nodes) {
    const int i = (int)(blockIdx.x * blockDim.x + threadIdx.x);
    if (i >= n_nodes * D) return;
    const int n = i / D;
    const int f = i - n * D;
    float dg = deg[n];
    dg = dg > 1.0f ? dg : 1.0f;
    aggh[i] = aggh[i] / dg + t[i] + b[f];
}

__global__ void zero_f32(float* __restrict__ p, int n) {
    const int i = (int)(blockIdx.x * blockDim.x + threadIdx.x);
    if (i < n) p[i] = 0.0f;
}

// ---------------------------------------------------------------------------
// Edge head: e = h2[src] * h2[dst];  sigmoid((e @ fc1 + b1) @ fc2 + b2)
// One thread per edge; h2 rows loaded as float4 (global_load_b128);
// fc weights are wave-uniform -> scalar loads / constant-cache hits.
// ---------------------------------------------------------------------------
__global__ void edge_head(const float* __restrict__ h2,
                          const long long* __restrict__ src,
                          const long long* __restrict__ dst,
                          const float* __restrict__ fc1w,   // [16][8]
                          const float* __restrict__ fc1b,   // [8]
                          const float* __restrict__ fc2w,   // [8]
                          const float* __restrict__ fc2b,   // [1]
                          float* __restrict__ out,
                          int n_edges) {
    const int e = (int)(blockIdx.x * blockDim.x + threadIdx.x);
    if (e >= n_edges) return;
    const int s = (int)src[e];
    const int d = (int)dst[e];

    const float4* hs = (const float4*)(h2 + (size_t)s * 16);
    const float4* hd = (const float4*)(h2 + (size_t)d * 16);

    float ev[16];
    #pragma unroll
    for (int q = 0; q < 4; ++q) {
        const float4 a = hs[q];
        const float4 b = hd[q];
        ev[q * 4 + 0] = a.x * b.x;
        ev[q * 4 + 1] = a.y * b.y;
        ev[q * 4 + 2] = a.z * b.z;
        ev[q * 4 + 3] = a.w * b.w;
    }

    float acc[8];
    #pragma unroll
    for (int j = 0; j < 8; ++j) acc[j] = fc1b[j];
    #pragma unroll
    for (int k = 0; k < 16; ++k) {
        const float x = ev[k];
        #pragma unroll
        for (int j = 0; j < 8; ++j) acc[j] = fmaf(x, fc1w[k * 8 + j], acc[j]);
    }
    float o = fc2b[0];
    #pragma unroll
    for (int j = 0; j < 8; ++j) o = fmaf(acc[j], fc2w[j], o);

    out[e] = 1.0f / (1.0f + expf(-o));
}

// ---------------------------------------------------------------------------
extern "C" void kernel_launch(void* const* d_in, const int* in_sizes, int n_in,
                              void* d_out, int out_size, void* d_ws, size_t ws_size,
                              hipStream_t stream) {
    const float*     x      = (const float*)d_in[0];
    const long long* eidx   = (const long long*)d_in[1];   // jnp.int64 [2][E]
    const float*     w1l    = (const float*)d_in[2];
    const float*     w1r    = (const float*)d_in[3];
    const float*     b1     = (const float*)d_in[4];
    const float*     w2l    = (const float*)d_in[5];
    const float*     w2r    = (const float*)d_in[6];
    const float*     b2     = (const float*)d_in[7];
    const float*     fc1w   = (const float*)d_in[8];
    const float*     fc1b   = (const float*)d_in[9];
    const float*     fc2w   = (const float*)d_in[10];
    const float*     fc2b   = (const float*)d_in[11];
    float*           out    = (float*)d_out;

    const int n_nodes = in_sizes[0] / D_IN;
    const int n_edges = in_sizes[1] / 2;
    const long long* src = eidx;
    const long long* dst = eidx + n_edges;

    // workspace layout (floats); layer-2 buffers reuse the dead p1 region
    float* ws   = (float*)d_ws;
    const size_t N64 = (size_t)n_nodes * 64;
    const size_t N16 = (size_t)n_nodes * 16;
    float* p1   = ws;                 // [N][64]  x @ w1_l
    float* t1   = ws + N64;           // [N][64]  x @ w1_r
    float* agg1 = ws + 2 * N64;       // [N][64]  -> becomes h1 in place
    float* deg  = ws + 3 * N64;       // [N]
    float* p2   = ws;                 // [N][16]  (reuses p1 region)
    float* t2   = ws + N16;           // [N][16]
    float* agg2 = ws + 2 * N16;       // [N][16]  -> becomes h2 in place
    float* h1   = agg1;
    float* h2   = agg2;

    const int nRowTiles = (n_nodes + 15) / 16;
    const int BLK = 256;                       // 8 wave32 per block

    // ---- layer 1: transform (WMMA), then aggregate transformed features ----
    {
        const int tiles = nRowTiles * (D_H1 / 16);
        const int grid  = (tiles + (BLK / 32) - 1) / (BLK / 32);
        gemm_dual_wmma<D_IN, D_H1><<<grid, BLK, 0, stream>>>(
            x, w1l, w1r, p1, t1, n_nodes, nRowTiles);
    }
    {
        const int n = (int)(N64 + n_nodes);    // agg1 and deg are contiguous
        zero_f32<<<(n + BLK - 1) / BLK, BLK, 0, stream>>>(agg1, n);
    }
    {
        const int n = n_edges * 64;
        scatter_add64<<<(n + BLK - 1) / BLK, BLK, 0, stream>>>(
            p1, src, dst, agg1, deg, n_edges);
    }
    {
        const int n = n_nodes * 64;
        combine<D_H1><<<(n + BLK - 1) / BLK, BLK, 0, stream>>>(
            agg1, t1, deg, b1, n_nodes);
    }

    // ---- layer 2 ----
    {
        const int tiles = nRowTiles * (D_H2 / 16);
        const int grid  = (tiles + (BLK / 32) - 1) / (BLK / 32);
        gemm_dual_wmma<D_H1, D_H2><<<grid, BLK, 0, stream>>>(
            h1, w2l, w2r, p2, t2, n_nodes, nRowTiles);
    }
    {
        const int n = (int)N16;
        zero_f32<<<(n + BLK - 1) / BLK, BLK, 0, stream>>>(agg2, n);
    }
    {
        const int n = n_edges * 16;
        scatter_add16<<<(n + BLK - 1) / BLK, BLK, 0, stream>>>(
            p2, src, dst, agg2, n_edges);
    }
    {
        const int n = n_nodes * 16;
        combine<D_H2><<<(n + BLK - 1) / BLK, BLK, 0, stream>>>(
            agg2, t2, deg, b2, n_nodes);
    }

    // ---- edge MLP head ----
    edge_head<<<(n_edges + BLK - 1) / BLK, BLK, 0, stream>>>(
        h2, src, dst, fc1w, fc1b, fc2w, fc2b, out, n_edges);
}